// HungarianMatcher_73761768341609
// MI455X (gfx1250) — compile-verified
//
#include <hip/hip_runtime.h>
#include <stdint.h>

// Problem constants (from reference setup_inputs): bs=16, Q=900, C=91, T=4096
#define NCLS   91
#define CPAD   96          // class row padded to 96 floats (384 B, 16B-multiple)
#define TDIM   4096
#define NROWS  72          // rows per block chunk; 14400 / 72 = 200 chunks
#define BT     256         // threads per block, 8 wave32
#define TPT    4           // targets per thread (consecutive -> b128 stores)
#define TBLK   (BT * TPT)  // 1024 targets per block; TDIM/TBLK = 4 t-tiles

#define ALPHA_   0.25f
#define EPS_     1e-8f
#define W_CLASS_ 2.0f
#define W_BBOX_  5.0f
#define W_GIOU_  2.0f

typedef float v4f __attribute__((ext_vector_type(4)));
typedef int   v4i __attribute__((ext_vector_type(4)));

// ---------------------------------------------------------------------------
// CDNA5 async global -> LDS copy, 16 bytes per enabled lane (ASYNCcnt).
// ---------------------------------------------------------------------------
__device__ __forceinline__ void async_b128(uint32_t lds_addr, const void* gaddr) {
  asm volatile("global_load_async_to_lds_b128 %0, %1, off"
               :
               : "v"(lds_addr), "v"((uint64_t)(uintptr_t)gaddr)
               : "memory");
}

__device__ __forceinline__ void wait_asynccnt0() {
  asm volatile("s_wait_asynccnt 0" ::: "memory");
}

// ---------------------------------------------------------------------------
// Kernel 1: per-(row, class) focal cost table, scaled by W_CLASS, padded to 96.
// ---------------------------------------------------------------------------
__global__ __launch_bounds__(256) void focal_table_kernel(
    const float* __restrict__ logits, float* __restrict__ table, int total) {
  int i = blockIdx.x * 256 + threadIdx.x;
  if (i >= total) return;
  int c = i % CPAD;
  float v = 0.0f;
  if (c < NCLS) {
    int n = i / CPAD;
    float x = logits[n * NCLS + c];
    float p = 1.0f / (1.0f + expf(-x));          // sigmoid
    float om = 1.0f - p;
    float pos = ALPHA_ * om * om * (-logf(p + EPS_));          // gamma = 2
    float neg = (1.0f - ALPHA_) * p * p * (-log1pf(EPS_ - p)); // -log(1-p+eps)
    v = W_CLASS_ * (pos - neg);
  }
  table[i] = v;
}

// ---------------------------------------------------------------------------
// Kernel 2: cost matrix. Block = 256 threads x 4 consecutive targets each
// (1024 targets) x NROWS rows. Chunk's class rows + pred boxes are
// async-staged into LDS; output written as non-temporal b128 stores.
// ---------------------------------------------------------------------------
__global__ __launch_bounds__(BT) void cost_matrix_kernel(
    const float* __restrict__ pred_boxes,  // [N,4] cxcywh
    const int*   __restrict__ tgt_ids,     // [T]
    const float* __restrict__ tgt_bbox,    // [T,4] cxcywh
    const float* __restrict__ cls_table,   // [N,CPAD]
    float* __restrict__ out,               // [N,T]
    int N) {
  __shared__ __align__(16) float s_box[NROWS * 4];       // 1152 B
  __shared__ __align__(16) float s_cls[NROWS * CPAD];    // 27648 B

  const int tid  = threadIdx.x;
  const int t0   = blockIdx.x * TBLK + tid * TPT;        // 4 consecutive targets
  const int n0   = blockIdx.y * NROWS;
  const int rows = min(NROWS, N - n0);

  // ---- Phase A: async stage chunk data into LDS --------------------------
  {
    const uint32_t lds_box = (uint32_t)(uintptr_t)&s_box[0];
    const uint32_t lds_cls = (uint32_t)(uintptr_t)&s_cls[0];
    const char* gbox = (const char*)(pred_boxes + (size_t)n0 * 4);
    const char* gcls = (const char*)(cls_table + (size_t)n0 * CPAD);

    if (tid < rows)                                  // rows * 16 B
      async_b128(lds_box + tid * 16, gbox + tid * 16);

    const int n16 = rows * (CPAD * 4 / 16);          // rows * 24 x b128
    for (int idx = tid; idx < n16; idx += BT)
      async_b128(lds_cls + idx * 16, gcls + idx * 16);
  }

  // ---- Per-thread targets into registers (overlaps with async fill) ------
  v4f tb[TPT];
  float tx0[TPT], ty0[TPT], tx1[TPT], ty1[TPT], areaT[TPT];
  int   id[TPT];
  {
    const v4i ids = ((const v4i*)tgt_ids)[t0 / 4];   // t0 is a multiple of 4
#pragma unroll
    for (int j = 0; j < TPT; ++j) {
      tb[j]    = ((const v4f*)tgt_bbox)[t0 + j];
      tx0[j]   = tb[j].x - 0.5f * tb[j].z;
      ty0[j]   = tb[j].y - 0.5f * tb[j].w;
      tx1[j]   = tb[j].x + 0.5f * tb[j].z;
      ty1[j]   = tb[j].y + 0.5f * tb[j].w;
      areaT[j] = tb[j].z * tb[j].w;
      id[j]    = ids[j];
    }
  }

  wait_asynccnt0();
  __syncthreads();

  float* outp = out + (size_t)n0 * TDIM + t0;

  for (int r = 0; r < rows; ++r) {
    const v4f ob = ((const v4f*)s_box)[r];           // ds broadcast
    // row-invariant conversion, amortized over 4 targets
    const float ox0 = ob.x - 0.5f * ob.z, oy0 = ob.y - 0.5f * ob.w;
    const float ox1 = ob.x + 0.5f * ob.z, oy1 = ob.y + 0.5f * ob.w;
    const float areaO = ob.z * ob.w;

    v4f res;
#pragma unroll
    for (int j = 0; j < TPT; ++j) {
      const float cls = s_cls[r * CPAD + id[j]];     // LDS gather

      const float l1 = fabsf(ob.x - tb[j].x) + fabsf(ob.y - tb[j].y) +
                       fabsf(ob.z - tb[j].z) + fabsf(ob.w - tb[j].w);

      float iw = fminf(ox1, tx1[j]) - fmaxf(ox0, tx0[j]); iw = fmaxf(iw, 0.0f);
      float ih = fminf(oy1, ty1[j]) - fmaxf(oy0, ty0[j]); ih = fmaxf(ih, 0.0f);
      const float inter = iw * ih;
      const float uni   = areaO + areaT[j] - inter;
      const float iou   = inter * __builtin_amdgcn_rcpf(uni);

      float ew = fmaxf(ox1, tx1[j]) - fminf(ox0, tx0[j]); ew = fmaxf(ew, 0.0f);
      float eh = fmaxf(oy1, ty1[j]) - fminf(oy0, ty0[j]); eh = fmaxf(eh, 0.0f);
      const float areaE = ew * eh;
      const float giou  = iou - (areaE - uni) * __builtin_amdgcn_rcpf(areaE);

      res[j] = W_BBOX_ * l1 + cls - W_GIOU_ * giou;
    }
    // write-once 236MB stream > 192MB L2: non-temporal b128 store
    __builtin_nontemporal_store(res, (v4f*)(outp + (size_t)r * TDIM));
  }
}

// ---------------------------------------------------------------------------
extern "C" void kernel_launch(void* const* d_in, const int* in_sizes, int n_in,
                              void* d_out, int out_size, void* d_ws, size_t ws_size,
                              hipStream_t stream) {
  const float* pred_logits = (const float*)d_in[0];  // [16,900,91]
  const float* pred_boxes  = (const float*)d_in[1];  // [16,900,4]
  const int*   tgt_ids     = (const int*)d_in[2];    // [4096]
  const float* tgt_bbox    = (const float*)d_in[3];  // [4096,4]
  float* out = (float*)d_out;

  const int N = in_sizes[1] / 4;                     // 14400
  float* cls_table = (float*)d_ws;                   // N * CPAD floats (~5.5 MB)

  const int total = N * CPAD;
  focal_table_kernel<<<(total + 255) / 256, 256, 0, stream>>>(
      pred_logits, cls_table, total);

  dim3 grid(TDIM / TBLK, (N + NROWS - 1) / NROWS);   // (4, 200)
  cost_matrix_kernel<<<grid, BT, 0, stream>>>(
      pred_boxes, tgt_ids, tgt_bbox, cls_table, out, N);
}